// GMMCovHead_52012053954891
// MI455X (gfx1250) — compile-verified
//
#include <hip/hip_runtime.h>
#include <hip/hip_bf16.h>

#define KG 8            // gaussians
#define DT 16           // dims
#define NCHP 136        // chol params per gaussian
#define SROW 1224       // pred row stride = 16*8 + 136*8 + 8
#define LOG2PID 29.40603f

typedef float v2f __attribute__((ext_vector_type(2)));
typedef float v4f __attribute__((ext_vector_type(4)));
typedef float v8f __attribute__((ext_vector_type(8)));

__global__ void __launch_bounds__(32) gmm_ws_init(float* ws) {
  if (threadIdx.x < 2) ws[threadIdx.x] = 0.0f;
}

__global__ void __launch_bounds__(256, 1) gmm_nll_wmma(
    const float* __restrict__ pred, const float* __restrict__ target,
    const float* __restrict__ mask, float* __restrict__ ws, int n) {
  const int lane = threadIdx.x & 31;
  const int wave = threadIdx.x >> 5;
  const int tl   = lane & 15;   // token within tile (WMMA row M)
  const int h    = lane >> 4;   // half selects which K-pair of the A slice
  const int t    = blockIdx.x * 128 + wave * 16 + tl;
  const int tc   = (t < n) ? t : (n - 1);   // clamp tail, keep EXEC full

  const float* row  = pred + (long long)tc * SROW;
  const float* trow = target + (long long)tc * DT;

  // Per-lane diag offsets (within row) and target values for this lane's 8 dims.
  int   dof[8];
  float tg[8];
#pragma unroll
  for (int j = 0; j < 4; ++j) {
    const int d0 = 4 * j + 2 * h;
    const int d1 = d0 + 1;
    dof[2 * j]     = DT * KG + d0 * (d0 + 3) / 2;  // 128 + tril diag index
    dof[2 * j + 1] = DT * KG + d1 * (d1 + 3) / 2;
    v2f tv = *(const v2f*)(trow + d0);
    tg[2 * j] = tv[0]; tg[2 * j + 1] = tv[1];
  }

  const v2f bones = {1.0f, 1.0f};  // all-ones B => D columns = row-sums of A
  float S[KG];                     // S[k] = sum_d( log(diag) + 0.5*y^2 )

#pragma unroll
  for (int k = 0; k < KG; ++k) {
    v8f acc = {0.f, 0.f, 0.f, 0.f, 0.f, 0.f, 0.f, 0.f};
#pragma unroll
    for (int j = 0; j < 4; ++j) {
      const int d0 = 4 * j + 2 * h;
      v2f mv = *(const v2f*)(row + k * DT + d0);
      float c0 = row[k * NCHP + dof[2 * j]];
      float c1 = row[k * NCHP + dof[2 * j + 1]];
      float g0 = fmaxf(__expf(c0), 1e-5f);
      float g1 = fmaxf(__expf(c1), 1e-5f);
      float i0 = __builtin_amdgcn_rcpf(g0);
      float i1 = __builtin_amdgcn_rcpf(g1);
      float y0 = (tg[2 * j]     - mv[0]) * i0;
      float y1 = (tg[2 * j + 1] - mv[1]) * i1;
      v2f a;
      a[0] = __logf(g0) + 0.5f * y0 * y0;
      a[1] = __logf(g1) + 0.5f * y1 * y1;
      // 16x16x4 f32 WMMA: D = A(16x4) * ones(4x16) + C  -> row sums, chained C
      acc = __builtin_amdgcn_wmma_f32_16x16x4_f32(
          false, a, false, bones, (short)0, acc, false, false);
    }
    // Extract per-token row sum from C/D layout:
    // D[m][n]: vgpr (m%8), lanes (n + 16*(m/8)); all columns equal.
    const int r = lane & 7;
    float s = acc[0];
    s = (r == 1) ? acc[1] : s;
    s = (r == 2) ? acc[2] : s;
    s = (r == 3) ? acc[3] : s;
    s = (r == 4) ? acc[4] : s;
    s = (r == 5) ? acc[5] : s;
    s = (r == 6) ? acc[6] : s;
    s = (r == 7) ? acc[7] : s;
    const int src = (lane & 8) ? (lane + 8) : lane;  // tokens 8..15 live in lanes 16..23
    S[k] = __shfl(s, src, 32);
  }

  // Logits: 8 contiguous floats, 16B-aligned.
  v4f l0 = *(const v4f*)(row + DT * KG + NCHP * KG);
  v4f l1 = *(const v4f*)(row + DT * KG + NCHP * KG + 4);
  float lo[KG] = {l0[0], l0[1], l0[2], l0[3], l1[0], l1[1], l1[2], l1[3]};

  // LSE(logits)
  float mw = lo[0];
#pragma unroll
  for (int k = 1; k < KG; ++k) mw = fmaxf(mw, lo[k]);
  float sw = 0.0f;
#pragma unroll
  for (int k = 0; k < KG; ++k) sw += __expf(lo[k] - mw);
  const float lse_w = mw + __logf(sw);

  // LSE_k(logit_k + log_prob_k), log_prob_k = -0.5*LOG2PID - S[k]
  float av[KG];
  float ma = -3.4e38f;
#pragma unroll
  for (int k = 0; k < KG; ++k) {
    av[k] = lo[k] - 0.5f * LOG2PID - S[k];
    ma = fmaxf(ma, av[k]);
  }
  float sa = 0.0f;
#pragma unroll
  for (int k = 0; k < KG; ++k) sa += __expf(av[k] - ma);
  const float lse_a = ma + __logf(sa);

  const float nll = lse_w - lse_a;
  const float mk  = mask[tc];
  const bool valid = (t < n) && (h == 0);      // one contribution per token
  float p0 = valid ? nll * mk : 0.0f;          // select (not multiply): no NaN*0
  float p1 = valid ? mk : 0.0f;

#pragma unroll
  for (int off = 16; off > 0; off >>= 1) {
    p0 += __shfl_xor(p0, off, 32);
    p1 += __shfl_xor(p1, off, 32);
  }
  if (lane == 0) {
    atomicAdd(&ws[0], p0);
    atomicAdd(&ws[1], p1);
  }
}

__global__ void __launch_bounds__(32) gmm_finalize(const float* ws, float* out) {
  if (threadIdx.x == 0) out[0] = ws[0] / ws[1];
}

extern "C" void kernel_launch(void* const* d_in, const int* in_sizes, int n_in,
                              void* d_out, int out_size, void* d_ws, size_t ws_size,
                              hipStream_t stream) {
  (void)n_in; (void)out_size; (void)ws_size;
  const float* pred   = (const float*)d_in[0];
  const float* target = (const float*)d_in[1];
  const float* mask   = (const float*)d_in[2];
  float* out = (float*)d_out;
  float* ws  = (float*)d_ws;

  const int n = in_sizes[2];              // number of tokens (mask length)
  const int grid = (n + 127) / 128;       // 128 tokens per 256-thread block

  gmm_ws_init<<<1, 32, 0, stream>>>(ws);
  gmm_nll_wmma<<<grid, 256, 0, stream>>>(pred, target, mask, ws, n);
  gmm_finalize<<<1, 32, 0, stream>>>(ws, out);
}